// GCN_motif_23192823399156
// MI455X (gfx1250) — compile-verified
//
#include <hip/hip_runtime.h>
#include <hip/hip_bf16.h>

// ---------------------------------------------------------------------------
// GCN 2-layer forward for MI455X (gfx1250, wave32).
//   layer: norm -> GEMM (fp32 WMMA 16x16x4) -> edge scatter (f32 atomics, L2
//   resident) -> self-loop + bias (+relu+dropout for layer 1).
// ---------------------------------------------------------------------------

#define NN   50000
#define EE   800000
#define C_IN 128
#define C_H  256
#define C_OUT 128

typedef __attribute__((ext_vector_type(2))) float v2f;
typedef __attribute__((ext_vector_type(8))) float v8f;

// ---------------- utility kernels ----------------

__global__ void fill_f32(float* __restrict__ p, float v, int n) {
  int i = blockIdx.x * blockDim.x + threadIdx.x;
  if (i < n) p[i] = v;
}

// deg[d] += w[e]  (deg pre-filled with 1.0 for self loops)
__global__ void deg_scatter(const long long* __restrict__ dst,
                            const float* __restrict__ w,
                            float* __restrict__ deg, int e_count) {
  int e = blockIdx.x * blockDim.x + threadIdx.x;
  if (e < e_count) atomicAdd(&deg[(int)dst[e]], w[e]);
}

// deg -> dinv in place
__global__ void make_dinv(float* __restrict__ deg, int n) {
  int i = blockIdx.x * blockDim.x + threadIdx.x;
  if (i < n) {
    float d = deg[i];
    deg[i] = (d > 0.0f) ? rsqrtf(d) : 0.0f;
  }
}

// norm[e] = dinv[s] * w[e] * dinv[d]
__global__ void make_norm(const long long* __restrict__ src,
                          const long long* __restrict__ dst,
                          const float* __restrict__ w,
                          const float* __restrict__ dinv,
                          float* __restrict__ norm, int e_count) {
  int e = blockIdx.x * blockDim.x + threadIdx.x;
  if (e < e_count) norm[e] = dinv[(int)src[e]] * w[e] * dinv[(int)dst[e]];
}

// ---------------- fp32 WMMA GEMM: Out[M,Nn] = X[M,K] @ W[K,Nn] -------------
// One wave per 16x64 output strip.  V_WMMA_F32_16X16X4_F32 per ISA layout:
//   A (16x4):  lane l holds (m = l&15, k = 2*(l>>4)+v)   in vgpr v
//   B (4x16):  lane l holds (k = 2*(l>>4)+v, n = l&15)   in vgpr v
//   D (16x16): lane l, vgpr r holds (m = r + 8*(l>>4), n = l&15)
__global__ void __launch_bounds__(32)
gemm_wmma_f32(const float* __restrict__ X, const float* __restrict__ W,
              float* __restrict__ Out, int M, int K, int Nn) {
  const int lane = threadIdx.x & 31;
  const int lo = lane & 15;
  const int hi = lane >> 4;

  const int ntiles = Nn >> 6;            // 64-wide super tiles
  const int bt = blockIdx.x;
  const int tm = (bt / ntiles) << 4;
  const int tn = (bt % ntiles) << 6;

  v8f acc0 = {}, acc1 = {}, acc2 = {}, acc3 = {};

  const float* __restrict__ xp = X + (size_t)(tm + lo) * K + 2 * hi;

  for (int k = 0; k < K; k += 4) {
    v2f a;
    a.x = xp[k];
    a.y = xp[k + 1];

    const float* __restrict__ wr0 = W + (size_t)(k + 2 * hi) * Nn + tn + lo;
    const float* __restrict__ wr1 = wr0 + Nn;
    v2f b0, b1, b2, b3;
    b0.x = wr0[0];  b0.y = wr1[0];
    b1.x = wr0[16]; b1.y = wr1[16];
    b2.x = wr0[32]; b2.y = wr1[32];
    b3.x = wr0[48]; b3.y = wr1[48];

    acc0 = __builtin_amdgcn_wmma_f32_16x16x4_f32(false, a, false, b0, (short)0, acc0, false, false);
    acc1 = __builtin_amdgcn_wmma_f32_16x16x4_f32(false, a, false, b1, (short)0, acc1, false, false);
    acc2 = __builtin_amdgcn_wmma_f32_16x16x4_f32(false, a, false, b2, (short)0, acc2, false, false);
    acc3 = __builtin_amdgcn_wmma_f32_16x16x4_f32(false, a, false, b3, (short)0, acc3, false, false);
  }

  float* __restrict__ orow = Out + (size_t)tm * Nn + tn + lo;
#pragma unroll
  for (int r = 0; r < 8; ++r) {
    const size_t row = (size_t)(r + 8 * hi) * Nn;
    orow[row + 0]  = acc0[r];
    orow[row + 16] = acc1[r];
    orow[row + 32] = acc2[r];
    orow[row + 48] = acc3[r];
  }
}

// ---------------- edge scatter: out[d,:] += H[s,:] * norm[e] ----------------
// blockDim.x == C (channels fully coalesced); a few edges per block.
template <int C>
__global__ void spmm_scatter(const long long* __restrict__ src,
                             const long long* __restrict__ dst,
                             const float* __restrict__ norm,
                             const float* __restrict__ H,
                             float* __restrict__ out,
                             int e_count, int edges_per_block) {
  const int c = threadIdx.x;
  int e0 = blockIdx.x * edges_per_block;
  int e1 = e0 + edges_per_block;
  if (e1 > e_count) e1 = e_count;
  for (int e = e0; e < e1; ++e) {
    const int s = (int)src[e];
    const int d = (int)dst[e];
    const float v = H[(size_t)s * C + c] * norm[e];
    atomicAdd(&out[(size_t)d * C + c], v);
  }
}

// ---------------- layer-1 finish: self-loop + bias + relu + dropout --------
__device__ __forceinline__ unsigned hash_u32(unsigned x) {
  x ^= x >> 16; x *= 0x7feb352dU;
  x ^= x >> 15; x *= 0x846ca68bU;
  x ^= x >> 16;
  return x;
}

__global__ void finish1(const float* __restrict__ H,
                        float* __restrict__ A,        // agg in, activation out
                        const float* __restrict__ dinv,
                        const float* __restrict__ b, int n, int C) {
  int idx = blockIdx.x * blockDim.x + threadIdx.x;
  if (idx >= n * C) return;
  const int i = idx / C;
  const int c = idx - i * C;
  const float di = dinv[i];
  float v = A[idx] + H[idx] * di * di + b[c];
  v = fmaxf(v, 0.0f);                               // relu
  const unsigned h = hash_u32((unsigned)idx * 2654435761u ^ 42u);
  A[idx] = (h & 1u) ? (v * 2.0f) : 0.0f;            // dropout p=0.5, scale 1/(1-p)
}

// ---------------- layer-2 finish: self-loop + bias -> d_out ----------------
__global__ void finish2(const float* __restrict__ H2,
                        float* __restrict__ out,      // agg in, result out
                        const float* __restrict__ dinv,
                        const float* __restrict__ b, int n, int C) {
  int idx = blockIdx.x * blockDim.x + threadIdx.x;
  if (idx >= n * C) return;
  const int i = idx / C;
  const int c = idx - i * C;
  const float di = dinv[i];
  out[idx] = out[idx] + H2[idx] * di * di + b[c];
}

// ---------------------------------------------------------------------------

extern "C" void kernel_launch(void* const* d_in, const int* in_sizes, int n_in,
                              void* d_out, int out_size, void* d_ws, size_t ws_size,
                              hipStream_t stream) {
  (void)in_sizes; (void)n_in; (void)out_size; (void)ws_size;

  const float*     x   = (const float*)d_in[0];            // [N, 128]
  const long long* ei  = (const long long*)d_in[1];        // [2, E] int64
  const float*     w   = (const float*)d_in[2];            // [E]
  const float*     W1  = (const float*)d_in[3];            // [128, 256]
  const float*     b1  = (const float*)d_in[4];            // [256]
  const float*     W2  = (const float*)d_in[5];            // [256, 128]
  const float*     b2  = (const float*)d_in[6];            // [128]
  float*           out = (float*)d_out;                    // [N, 128]

  const long long* src = ei;
  const long long* dst = ei + EE;

  // workspace layout (floats)
  float* ws   = (float*)d_ws;
  float* dinv = ws;                                        // N
  float* norm = dinv + NN;                                 // E
  float* H    = norm + EE;                                 // N*256 (also H2 reuse)
  float* A    = H + (size_t)NN * C_H;                      // N*256
  float* H2   = H;                                         // reuse after finish1

  const int TB = 256;

  // ---- normalization coefficients (shared by both layers) ----
  fill_f32<<<(NN + TB - 1) / TB, TB, 0, stream>>>(dinv, 1.0f, NN);   // self loops
  deg_scatter<<<(EE + TB - 1) / TB, TB, 0, stream>>>(dst, w, dinv, EE);
  make_dinv<<<(NN + TB - 1) / TB, TB, 0, stream>>>(dinv, NN);
  make_norm<<<(EE + TB - 1) / TB, TB, 0, stream>>>(src, dst, w, dinv, norm, EE);

  // ---- layer 1: H = x @ W1 ----
  gemm_wmma_f32<<<(NN / 16) * (C_H / 64), 32, 0, stream>>>(x, W1, H, NN, C_IN, C_H);

  // ---- layer 1 aggregation ----
  fill_f32<<<((NN * C_H) + TB - 1) / TB, TB, 0, stream>>>(A, 0.0f, NN * C_H);
  {
    const int epb = 4;
    spmm_scatter<C_H><<<(EE + epb - 1) / epb, C_H, 0, stream>>>(src, dst, norm, H, A, EE, epb);
  }
  finish1<<<((NN * C_H) + TB - 1) / TB, TB, 0, stream>>>(H, A, dinv, b1, NN, C_H);

  // ---- layer 2: H2 = A @ W2 ----
  gemm_wmma_f32<<<(NN / 16) * (C_OUT / 64), 32, 0, stream>>>(A, W2, H2, NN, C_H, C_OUT);

  // ---- layer 2 aggregation -> d_out ----
  fill_f32<<<((NN * C_OUT) + TB - 1) / TB, TB, 0, stream>>>(out, 0.0f, NN * C_OUT);
  {
    const int epb = 4;
    spmm_scatter<C_OUT><<<(EE + epb - 1) / epb, C_OUT, 0, stream>>>(src, dst, norm, H2, out, EE, epb);
  }
  finish2<<<((NN * C_OUT) + TB - 1) / TB, TB, 0, stream>>>(H2, out, dinv, b2, NN, C_OUT);
}